// AttentionBlock_51445118272039
// MI455X (gfx1250) — compile-verified
//
#include <hip/hip_runtime.h>
#include <math.h>

typedef float v2f __attribute__((ext_vector_type(2)));
typedef float v8f __attribute__((ext_vector_type(8)));

#define T_TOK 1024
#define NKV 8
#define QM 8
#define HD 64
#define WIN 128
#define SM_SCALE 0.125f   // 1/sqrt(64)

// RoPE'd element j of a 64-float row, given per-position cos/sin rows (32 each)
__device__ __forceinline__ float ropeVal(const float* __restrict__ row,
                                         const float* __restrict__ cR,
                                         const float* __restrict__ sR, int j) {
    if (j < 32) return row[j] * cR[j] - row[j + 32] * sR[j];
    int i = j - 32;
    return row[j] * cR[i] + row[i] * sR[i];
}

__global__ __launch_bounds__(256)
void attn_swa_sink_kernel(const float* __restrict__ Q, const float* __restrict__ K,
                          const float* __restrict__ V, const float* __restrict__ S,
                          const float* __restrict__ cosT, const float* __restrict__ sinT,
                          float* __restrict__ out)
{
    __shared__ float sK[16 * HD];        // roped K tile  [key][d]
    __shared__ float sV[16 * HD];        // V tile        [key][d]
    __shared__ float sP[QM][16 * 18];    // per-wave P tile, padded stride 18

    const int qt   = blockIdx.x;         // query tile (16 rows)
    const int h    = blockIdx.y;         // kv head
    const int m    = threadIdx.y;        // q-head within kv head (one wave each)
    const int lane = threadIdx.x;        // 0..31
    const int tid  = threadIdx.y * 32 + lane;
    const int q0   = qt * 16;
    const int half = lane >> 4;          // 0: lanes 0-15, 1: lanes 16-31
    const int l16  = lane & 15;

    // ---- Q A-fragments (16x4 fp32 layout), RoPE + sm_scale folded in ----
    // A layout: lane L(<16) holds row L, K = {4c, 4c+1}; lane L+16 holds row L, K = {4c+2, 4c+3}
    v2f qa[16];
    {
        const int qrow = q0 + l16;
        const float* qptr = Q + (((size_t)qrow * NKV + h) * QM + m) * HD;
        const float* cR = cosT + (size_t)qrow * 32;
        const float* sR = sinT + (size_t)qrow * 32;
#pragma unroll
        for (int c = 0; c < 16; ++c) {
            const int j = 4 * c + 2 * half;
            qa[c].x = ropeVal(qptr, cR, sR, j)     * SM_SCALE;
            qa[c].y = ropeVal(qptr, cR, sR, j + 1) * SM_SCALE;
        }
    }

    // ---- softmax state: sink folded into init (m = sink, l = 1) ----
    const float sink = S[h * QM + m];
    float mx[8], lsum[8];
    v8f acc[4];
#pragma unroll
    for (int r = 0; r < 8; ++r) { mx[r] = sink; lsum[r] = 1.0f; }
#pragma unroll
    for (int nb = 0; nb < 4; ++nb)
#pragma unroll
        for (int r = 0; r < 8; ++r) acc[nb][r] = 0.0f;

    const int kt0 = (qt > 8) ? (qt - 8) : 0;   // window is tile-aligned
    for (int kt = kt0; kt <= qt; ++kt) {
        const int k0 = kt * 16;

        __syncthreads();   // previous iteration's sK/sV consumers are done
        // ---- stage roped K tile + V tile: 1024 floats each, 256 threads x 4 ----
#pragma unroll
        for (int i = 0; i < 4; ++i) {
            const int idx = tid + i * 256;
            const int n = idx >> 6;        // key within tile
            const int d = idx & 63;
            const int kp = k0 + n;         // always in [0, T)
            const float* kptr = K + ((size_t)kp * NKV + h) * HD;
            const float* vptr = V + ((size_t)kp * NKV + h) * HD;
            const float* cR = cosT + (size_t)kp * 32;
            const float* sR = sinT + (size_t)kp * 32;
            sK[idx] = ropeVal(kptr, cR, sR, d);
            sV[idx] = vptr[d];
        }
        __syncthreads();

        // ---- GEMM1: Stile = Qr * Kr^T  (16x16, K-dim 64 in 16 chunks of 4) ----
        // B layout (4x16): lane n(<16): B[4c+0][n], B[4c+1][n]; lane n+16: B[4c+2][n], B[4c+3][n]
        // B[d][key] = Kr[key][d] -> float2 at sK[key*64 + 4c + 2*half]
        v8f sacc = {0.f, 0.f, 0.f, 0.f, 0.f, 0.f, 0.f, 0.f};
#pragma unroll
        for (int c = 0; c < 16; ++c) {
            const float* bp = &sK[l16 * HD + 4 * c + 2 * half];
            v2f b; b.x = bp[0]; b.y = bp[1];
            sacc = __builtin_amdgcn_wmma_f32_16x16x4_f32(
                false, qa[c], false, b, (short)0, sacc, false, false);
        }

        // ---- mask + online softmax (D layout: lane holds col l16, rows r+8*half) ----
#pragma unroll
        for (int r = 0; r < 8; ++r) {
            const int M = r + 8 * half;
            const int q = q0 + M;
            const int k = k0 + l16;
            float s = sacc[r];
            const bool ok = (k <= q) && (q - k <= WIN);
            s = ok ? s : -3.0e38f;

            float rm = s;
#pragma unroll
            for (int off = 1; off < 16; off <<= 1)
                rm = fmaxf(rm, __shfl_xor(rm, off, 32));
            const float mnew = fmaxf(mx[r], rm);
            const float p = __expf(s - mnew);
            float rs = p;
#pragma unroll
            for (int off = 1; off < 16; off <<= 1)
                rs += __shfl_xor(rs, off, 32);
            const float alpha = __expf(mx[r] - mnew);
            mx[r] = mnew;
            lsum[r] = lsum[r] * alpha + rs;
#pragma unroll
            for (int nb = 0; nb < 4; ++nb) acc[nb][r] *= alpha;

            sP[m][M * 18 + l16] = p;   // D-layout -> LDS for A-layout reload
        }
        __syncthreads();

        // ---- GEMM2: O += P * V  (A = P 16x16, 4 K-chunks; B = V, 4 N-tiles) ----
        v2f ap[4];
#pragma unroll
        for (int c = 0; c < 4; ++c) {
            const float* pp = &sP[m][l16 * 18 + 4 * c + 2 * half];
            ap[c].x = pp[0]; ap[c].y = pp[1];
        }
#pragma unroll
        for (int nb = 0; nb < 4; ++nb) {
#pragma unroll
            for (int c = 0; c < 4; ++c) {
                const int krow = 4 * c + 2 * half;
                v2f b;
                b.x = sV[krow * HD + 16 * nb + l16];
                b.y = sV[(krow + 1) * HD + 16 * nb + l16];
                acc[nb] = __builtin_amdgcn_wmma_f32_16x16x4_f32(
                    false, ap[c], false, b, (short)0, acc[nb], false, false);
            }
        }
    }

    // ---- epilogue: divide by softmax denominator (sink weight already only in lsum) ----
#pragma unroll
    for (int r = 0; r < 8; ++r) {
        const int M = r + 8 * half;
        const int q = q0 + M;
        const float inv = 1.0f / lsum[r];
        float* op = out + (size_t)q * (NKV * QM * HD) + h * (QM * HD) + m * HD;
#pragma unroll
        for (int nb = 0; nb < 4; ++nb)
            op[16 * nb + l16] = acc[nb][r] * inv;
    }
}

extern "C" void kernel_launch(void* const* d_in, const int* in_sizes, int n_in,
                              void* d_out, int out_size, void* d_ws, size_t ws_size,
                              hipStream_t stream) {
    const float* Q    = (const float*)d_in[0];
    const float* K    = (const float*)d_in[1];
    const float* V    = (const float*)d_in[2];
    const float* S    = (const float*)d_in[3];
    const float* cosT = (const float*)d_in[4];
    const float* sinT = (const float*)d_in[5];
    float* out = (float*)d_out;

    dim3 grid(T_TOK / 16, NKV);   // 64 q-tiles x 8 kv heads
    dim3 block(32, QM);           // 8 waves; wave w = q-head w
    attn_swa_sink_kernel<<<grid, block, 0, stream>>>(Q, K, V, S, cosT, sinT, out);
}